// Consistent_loss_right_25288767439319
// MI455X (gfx1250) — compile-verified
//
#include <hip/hip_runtime.h>

typedef float v2f __attribute__((ext_vector_type(2)));
typedef float v8f __attribute__((ext_vector_type(8)));

#define THREADS 256
#define MAXBLOCKS 4096
#define SWIZZLE_SWAPX16 0x401F   // group-of-32: xor_mask=0x10, and_mask=0x1f

// Pass 1: masked-abs partial sums over `up`, one partial per block.
// Reference reduces to mean( where(|up|<0.2, |up|, 0) ): the scatter-min is a
// structural no-op (min(0, positive)==0 for any data), so only `up` matters.
__global__ void __launch_bounds__(THREADS)
consistent_loss_partial(const float* __restrict__ up,
                        float* __restrict__ partials, int n) {
    const float thr = 0.2f;
    const int tid    = blockIdx.x * blockDim.x + threadIdx.x;
    const int stride = gridDim.x * blockDim.x;
    const int n4     = n >> 2;
    const float4* __restrict__ up4 = (const float4*)up;

    // 4 independent accumulators to break the add dependency chain.
    float a0 = 0.f, a1 = 0.f, a2 = 0.f, a3 = 0.f;
    for (int i = tid; i < n4; i += stride) {
        // gfx1250 speculative prefetch (global_prefetch_b8); OOB is dropped.
        __builtin_prefetch(&up4[i + stride], 0, 0);
        float4 v = up4[i];
        float d0 = fabsf(v.x), d1 = fabsf(v.y), d2 = fabsf(v.z), d3 = fabsf(v.w);
        a0 += (d0 < thr) ? d0 : 0.0f;
        a1 += (d1 < thr) ? d1 : 0.0f;
        a2 += (d2 < thr) ? d2 : 0.0f;
        a3 += (d3 < thr) ? d3 : 0.0f;
    }
    float sa = a0 + a1;
    float sb = a2 + a3;
    // Scalar tail (n not divisible by 4); no-op for the reference shape.
    for (int i = (n4 << 2) + tid; i < n; i += stride) {
        float d = fabsf(up[i]);
        sa += (d < thr) ? d : 0.0f;
    }

    // ---- Wave32 reduction via V_WMMA_F32_16X16X4_F32 (exact f32 math) ----
    // A = per-lane pair {sa, sb} (16x4: lane m -> A[m,0..1], lane m+16 ->
    // A[m,2..3]); B = ones(4x16). Then D[m,n] = sa_m + sb_m + sa_{m+16} +
    // sb_{m+16} =: r_m, row-constant. In the C/D layout lane l holds
    // d[v] = r_{v + 8*(l>=16)}, so the in-lane sum of d[0..7] is the
    // half-wave total; one SWAPX16 swizzle merges the two halves.
    // EXEC is all-ones here (loops reconverged, full waves).
    v2f a = {sa, sb};
    v2f b = {1.0f, 1.0f};
    v8f c = {};
    v8f d = __builtin_amdgcn_wmma_f32_16x16x4_f32(
        /*neg_a=*/false, a, /*neg_b=*/false, b,
        /*c_mod=*/(short)0, c, /*reuse_a=*/false, /*reuse_b=*/false);
    float half = ((d[0] + d[1]) + (d[2] + d[3])) +
                 ((d[4] + d[5]) + (d[6] + d[7]));
    // ds_swizzle_b32 SWAPX16: exchange lanes l <-> l^16 (single DS op).
    float other = __int_as_float(
        __builtin_amdgcn_ds_swizzle(__float_as_int(half), SWIZZLE_SWAPX16));
    float wave_total = half + other;

    // ---- Cross-wave (8 waves/block): LDS, fixed-order sum by thread 0 ----
    __shared__ float wsum[THREADS / 32];
    const int lane = threadIdx.x & 31;
    const int wave = threadIdx.x >> 5;
    if (lane == 0) wsum[wave] = wave_total;
    __syncthreads();
    if (threadIdx.x == 0) {
        float4 p = ((float4*)wsum)[0];   // ds_load_b128
        float4 q = ((float4*)wsum)[1];   // ds_load_b128
        partials[blockIdx.x] =
            ((p.x + p.y) + (p.z + p.w)) + ((q.x + q.y) + (q.z + q.w));
    }
}

// Pass 2: deterministic tree-reduction of block partials, scale by 1/N.
__global__ void __launch_bounds__(THREADS)
consistent_loss_final(const float* __restrict__ partials, int nblocks,
                      float* __restrict__ out, float inv_n) {
    __shared__ float lds[THREADS];
    float s = 0.0f;
    for (int i = threadIdx.x; i < nblocks; i += THREADS) s += partials[i];
    lds[threadIdx.x] = s;
    __syncthreads();
    for (int off = THREADS / 2; off > 0; off >>= 1) {
        if ((int)threadIdx.x < off) lds[threadIdx.x] += lds[threadIdx.x + off];
        __syncthreads();
    }
    if (threadIdx.x == 0) out[0] = lds[0] * inv_n;
}

extern "C" void kernel_launch(void* const* d_in, const int* in_sizes, int n_in,
                              void* d_out, int out_size, void* d_ws, size_t ws_size,
                              hipStream_t stream) {
    // setup_inputs order: up, left, right. Output depends only on `up`.
    const float* up = (const float*)d_in[0];
    const int n = in_sizes[0];              // 64*1*512*512 = 16,777,216

    float* partials = (float*)d_ws;
    int nblocks = MAXBLOCKS;                // 4096 partials = 16 KB of d_ws
    const int maxb = (int)(ws_size / sizeof(float));
    if (nblocks > maxb) nblocks = maxb;
    if (nblocks < 1) nblocks = 1;

    consistent_loss_partial<<<nblocks, THREADS, 0, stream>>>(up, partials, n);
    consistent_loss_final<<<1, THREADS, 0, stream>>>(partials, nblocks,
                                                     (float*)d_out,
                                                     1.0f / (float)n);
}